// Decoder_27522150433510
// MI455X (gfx1250) — compile-verified
//
#include <hip/hip_runtime.h>

// ---------------------------------------------------------------------------
// MI455X (gfx1250) implementation.
//
// Roofline: ~76 GFLOP fp32 GEMM work, ~1.3 GB of traffic, all L2-resident
// (192 MB L2). Memory/L2-bound => keep fp32 precision and use
// V_WMMA_F32_16X16X4_F32 for every einsum GEMM (2048 FLOP/instr, wave32).
// Per-sample weights (129 KB) are DMA'd into LDS once per workgroup via the
// Tensor Data Mover. Weights are PAIR-INTERLEAVED ([k/2][m][2]) so each WMMA
// A-fragment is one aligned ds_load_b64, and TDM pads every 128 DWORDs by 32
// (pair-row stride 160 words) so the half-wave offset is +32 banks =>
// conflict-free. GEMM1's K loop is split at the state/grad channel boundary
// so the hot loop is branch-free.
// ---------------------------------------------------------------------------

#define FCH   64          // F
#define C3F   192         // 3F
#define HH    128         // H
#define NB    4           // B
#define HW    (HH*HH)     // 16384
#define HYP   33024       // hypernetwork output size per sample
#define NEGS  0.2f
#define EPS_  1e-5f

// packed per-sample layout (pair-interleaved matrices, biases in tail block)
#define P_WIN   0         // W_in  interleaved, 96 pair-rows x 128
#define P_WMID  12288     // W_mid interleaved, 32 pair-rows
#define P_WOUT  16384     // W_out interleaved, 32 pair-rows
#define P_WSH   20480     // W_sh  interleaved, 96 pair-rows
#define P_BIAS  32768     // b_in[64] | b_mid[64] | b_out[64] | b_sh[64]

// LDS padded pair-row mapping: element e -> (e>>7)*PPROW + (e&127)
#define PPROW   160       // 128 data + 32 pad DWORDs per pair-row
#define NPAIRS  (HYP/128) // 258 pair-rows
// pair-row base indices per matrix
#define PR_WIN   0
#define PR_WMID  96
#define PR_WOUT  128
#define PR_WSH   160
// bias LDS word offsets (bias block = pair-rows 256,257)
#define BI_IN   (256*PPROW)
#define BI_MID  (256*PPROW + 64)
#define BI_OUT  (257*PPROW)
#define BI_SH   (257*PPROW + 64)

typedef float v2f __attribute__((ext_vector_type(2)));
typedef float v8f __attribute__((ext_vector_type(8)));
typedef unsigned int v4u __attribute__((ext_vector_type(4)));
typedef int v4i __attribute__((ext_vector_type(4)));
typedef int v8i __attribute__((ext_vector_type(8)));

#if defined(__has_builtin)
#  if __has_builtin(__builtin_amdgcn_tensor_load_to_lds)
#    define HAVE_TDM 1
#  endif
#endif
#if defined(HAVE_TDM)
#  if __has_include(<hip/amd_detail/amd_gfx1250_TDM.h>)
#    define TDM_ARITY6 1
#  endif
#endif

// dynamic-LDS layout for dyna_step (floats)
#define L_WPAD  (NPAIRS*PPROW)            // 41280: padded weight region
#define L_MEAN  (L_WPAD)                  // 192
#define L_ISTD  (L_WPAD + C3F)            // 192
#define L_H     (L_WPAD + 2*C3F)          // 8*64*16 per-wave h buffers
#define L_TOTAL (L_WPAD + 2*C3F + 8*FCH*16)   // 49856 floats = 199,424 bytes

__device__ __forceinline__ float lrelu(float x) { return x >= 0.f ? x : NEGS * x; }

// ---------------------------------------------------------------------------
// Kernel 0: hypernetwork GEMM  ws[b,j] = lat[b,:]·hyper_w[j,:] + hyper_b[j]
// Writes each matrix pair-interleaved: element (o,c) -> (c>>1)*128 + o*2 + (c&1)
// so a WMMA A-fragment (K pair for one M) is 8 contiguous bytes in LDS.
// ---------------------------------------------------------------------------
__global__ __launch_bounds__(256) void hyper_kernel(
    const float* __restrict__ lat,      // [4][512]
    const float* __restrict__ hyper_w,  // [33024][512]
    const float* __restrict__ hyper_b,  // [33024]
    float* __restrict__ wt)             // [4][33024] (pair-interleaved)
{
  int j = blockIdx.x * 256 + threadIdx.x;
  if (j >= HYP) return;
  const float* wr = hyper_w + (size_t)j * 512;
  float a0 = 0.f, a1 = 0.f, a2 = 0.f, a3 = 0.f;
#pragma unroll 8
  for (int k = 0; k < 512; ++k) {
    float wv = wr[k];
    a0 += wv * lat[k];
    a1 += wv * lat[512 + k];
    a2 += wv * lat[1024 + k];
    a3 += wv * lat[1536 + k];
  }
  float bb = hyper_b[j];
  // reference layout offsets: w_in 0 | b_in 12288 | w_mid 12352 | b_mid 16448
  //                           w_out 16512 | b_out 20608 | w_sh 20672 | b_sh 32960
  int dst;
  if (j < 12288)      { int o = j / C3F, c = j % C3F;
                        dst = P_WIN  + ((c >> 1) << 7) + (o << 1) + (c & 1); }
  else if (j < 12352) { dst = P_BIAS + (j - 12288); }                 // b_in
  else if (j < 16448) { int t = j - 12352; int o = t >> 6, c = t & 63;
                        dst = P_WMID + ((c >> 1) << 7) + (o << 1) + (c & 1); }
  else if (j < 16512) { dst = P_BIAS + 64 + (j - 16448); }            // b_mid
  else if (j < 20608) { int t = j - 16512; int o = t >> 6, c = t & 63;
                        dst = P_WOUT + ((c >> 1) << 7) + (o << 1) + (c & 1); }
  else if (j < 20672) { dst = P_BIAS + 128 + (j - 20608); }           // b_out
  else if (j < 32960) { int t = j - 20672; int o = t / C3F, c = t % C3F;
                        dst = P_WSH  + ((c >> 1) << 7) + (o << 1) + (c & 1); }
  else                { dst = P_BIAS + 192 + (j - 32960); }           // b_sh
  wt[dst]            = a0 + bb;
  wt[HYP + dst]      = a1 + bb;
  wt[2 * HYP + dst]  = a2 + bb;
  wt[3 * HYP + dst]  = a3 + bb;
}

// ---------------------------------------------------------------------------
// Kernel 1: SinSobel depthwise 5x5 grads + InstanceNorm statistics.
// ---------------------------------------------------------------------------
__global__ __launch_bounds__(256) void sobel_stats_kernel(
    const float* __restrict__ state,   // [B][64][128][128]
    float* __restrict__ grad,          // [B][128][128][128]
    float* __restrict__ meanArr,       // [B][192]
    float* __restrict__ istdArr)       // [B][192]
{
  const int bf = blockIdx.x;           // b*64+f
  const int b = bf >> 6, f = bf & 63;
  const int tid = threadIdx.x;
  const float* img = state + (size_t)bf * HW;
  float* gx_out = grad + ((size_t)b * 2 * FCH + 2 * f) * HW;
  float* gy_out = gx_out + HW;

  __shared__ float tile[12 * 132];     // 8 rows + 4 halo, 132 cols
  __shared__ float red[256];

  const float S_[5] = {0.f, 0.70710678118654752f, 1.f, 0.70710678118654752f, 0.f};
  const float D_[5] = {-1.f, -0.70710678118654752f, 0.f, 0.70710678118654752f, 1.f};

  float sid = 0.f, sid2 = 0.f, sgx = 0.f, sgx2 = 0.f, sgy = 0.f, sgy2 = 0.f;

  for (int r0 = 0; r0 < HH; r0 += 8) {
    __syncthreads();
    for (int i = tid; i < 12 * 132; i += 256) {
      int tr = i / 132, tc = i % 132;
      int gr = r0 + tr - 2, gc = tc - 2;
      float v = 0.f;
      if (gr >= 0 && gr < HH && gc >= 0 && gc < HH) v = img[gr * HH + gc];
      tile[i] = v;
    }
    __syncthreads();
    for (int t = tid; t < 8 * HH; t += 256) {
      int rr = t >> 7, px = t & 127;
      float gx = 0.f, gy = 0.f;
#pragma unroll
      for (int i = 0; i < 5; ++i)
#pragma unroll
        for (int jj = 0; jj < 5; ++jj) {
          float v = tile[(rr + i) * 132 + px + jj];
          gx += v * (S_[i] * D_[jj]);
          gy += v * (D_[i] * S_[jj]);
        }
      float c = tile[(rr + 2) * 132 + px + 2];
      gx_out[(r0 + rr) * HH + px] = gx;
      gy_out[(r0 + rr) * HH + px] = gy;
      sid += c;  sid2 += c * c;
      sgx += gx; sgx2 += gx * gx;
      sgy += gy; sgy2 += gy * gy;
    }
  }

  auto blockReduce = [&](float v) -> float {
    __syncthreads();
    red[tid] = v;
    __syncthreads();
    for (int s = 128; s > 0; s >>= 1) {
      if (tid < s) red[tid] += red[tid + s];
      __syncthreads();
    }
    return red[0];
  };

  const float invN = 1.f / (float)HW;
  float m0 = blockReduce(sid)  * invN;
  float q0 = blockReduce(sid2) * invN;
  float m1 = blockReduce(sgx)  * invN;
  float q1 = blockReduce(sgx2) * invN;
  float m2 = blockReduce(sgy)  * invN;
  float q2 = blockReduce(sgy2) * invN;
  if (tid == 0) {
    int base = b * C3F;
    meanArr[base + f] = m0;
    istdArr[base + f] = rsqrtf(q0 - m0 * m0 + EPS_);
    meanArr[base + FCH + 2 * f] = m1;
    istdArr[base + FCH + 2 * f] = rsqrtf(q1 - m1 * m1 + EPS_);
    meanArr[base + FCH + 2 * f + 1] = m2;
    istdArr[base + FCH + 2 * f + 1] = rsqrtf(q2 - m2 * m2 + EPS_);
  }
}

// ---------------------------------------------------------------------------
// Kernel 2: fused normalize + DynaResidualBlock via V_WMMA_F32_16X16X4_F32.
// Grid (row=128, b=4); 256 threads = 8 wave32s; each wave owns 16 pixels and
// all 64 output channels (4 M-tiles). Per-sample weights are TDM-loaded into
// LDS with destination padding (128->160 DWORD pair-rows, bank-conflict-free
// b64 A-fragment loads).
//
// Fragment lane layouts (wave32, per ISA 7.12.2):
//   A 16x4 f32 : lanes 0-15 M=lane, v.x=K0 v.y=K1 ; lanes 16-31 v.x=K2 v.y=K3
//   B 4x16 f32 : assumed mirror: lanes 0-15 N=lane, v.x=K0 v.y=K1 ;
//                lanes 16-31 v.x=K2 v.y=K3
//   C/D 16x16  : VGPR r, lanes 0-15 -> M=r, lanes 16-31 -> M=r+8, N=lane&15
// ---------------------------------------------------------------------------
__global__ __launch_bounds__(256) void dyna_step_kernel(
    const float* __restrict__ state_in,   // [B][64][128][128]
    const float* __restrict__ grad,       // [B][128][128][128]
    const float* __restrict__ meanArr,    // [B][192]
    const float* __restrict__ istdArr,    // [B][192]
    const float* __restrict__ wt,         // [B][33024] pair-interleaved
    const float* __restrict__ lf_ptr,
    float* __restrict__ state_out)
{
  const int row  = blockIdx.x;
  const int b    = blockIdx.y;
  const int tid  = threadIdx.x;
  const int wave = tid >> 5;
  const int lane = tid & 31;
  const int half = lane >> 4;
  const int ll   = lane & 15;
  const int pix  = wave * 16 + ll;

  extern __shared__ float smem[];
  float* sW    = smem;            // padded weight region: pair-row p at p*PPROW
  float* sMean = smem + L_MEAN;
  float* sIstd = smem + L_ISTD;
  float* sH    = smem + L_H;

  const float* wb = wt + (size_t)b * HYP;

  // stage norm params (all waves, overlaps with TDM weight DMA below)
  for (int i = tid; i < C3F; i += 256) {
    sMean[i] = meanArr[b * C3F + i];
    sIstd[i] = istdArr[b * C3F + i];
  }

#if defined(HAVE_TDM)
  if (wave == 0) {
    // Tensor DMA descriptor (CDNA5 ISA):
    // 1-row 2D tensor, tile_dim0 = tensor_dim0 = 33024 f32 elements -> LDS,
    // with pad_enable: +32 DWORDs after every 128 DWORDs (pair-row stride 160).
    unsigned long long ga = (unsigned long long)(const void*)wb;
    unsigned int lds_b = (unsigned int)(unsigned long long)(void*)sW;
    v4u g0;
    g0.x = 1u;                                   // count=1, user descriptor
    g0.y = lds_b;                                // lds_addr (bytes)
    g0.z = (unsigned int)ga;                     // global_addr[31:0]
    g0.w = (unsigned int)((ga >> 32) & 0x1FFFFFFull) | 0x80000000u; // addr[56:32] | type=2
    v8i g1;
    g1[0] = (int)(0x20000u                       // data_size = 4B
                  | (1u << 20)                   // pad_enable
                  | (6u << 22)                   // pad_interval: 2^(6+1) = 128 DWORDs
                  | (31u << 25));                // pad_amount: 31+1 = 32 DWORDs
    g1[1] = (int)0x81000000u;                    // tensor_dim0.lo16 (33024) << 16
    g1[2] = 0x00010000;                          // tensor_dim0.hi=0 | tensor_dim1.lo16=1
    g1[3] = (int)0x81000000u;                    // tensor_dim1.hi=0 | tile_dim0=33024
    g1[4] = 1;                                   // tile_dim1=1, tile_dim2=0
    g1[5] = HYP;                                 // tensor_dim0_stride
    g1[6] = 0;
    g1[7] = 0;
    v4i z4 = {};
#if defined(TDM_ARITY6)
    v8i z8 = {};
    __builtin_amdgcn_tensor_load_to_lds(g0, g1, z4, z4, z8, 0);
#else
    __builtin_amdgcn_tensor_load_to_lds(g0, g1, z4, z4, 0);
#endif
    __builtin_amdgcn_s_wait_tensorcnt(0);
  }
#else
  // fallback: cooperative copy with the same 128->160 DWORD pair-row padding
  for (int i = tid; i < HYP; i += 256)
    sW[(i >> 7) * PPROW + (i & 127)] = wb[i];
#endif
  __syncthreads();

  const size_t identBase = (size_t)b * FCH * HW + (size_t)row * HH;
  const size_t gradBase  = (size_t)b * 2 * FCH * HW + (size_t)row * HH;
  // per-lane base pointers pre-offset by the half-wave K offset (branch-free)
  const float* pState = state_in + identBase + (size_t)(2 * half) * HW + pix;
  const float* pGrad  = grad + gradBase + (size_t)(2 * half) * HW + pix;

  const int aoff = (ll << 1);   // lane's element offset within a pair-row block
  v8f accH[4] = {};
  v8f accS[4] = {};

  // ---- GEMM1a: K in [0,64) from state channels --------------------------
#pragma unroll 4
  for (int kb = 0; kb < FCH; kb += 4) {
    const int k0 = kb + 2 * half;
    const int p0 = (kb >> 1) + half;           // pair-row for (k0, k0+1)
    v2f bf;
    bf.x = (pState[(size_t)kb * HW]      - sMean[k0])     * sIstd[k0];
    bf.y = (pState[(size_t)kb * HW + HW] - sMean[k0 + 1]) * sIstd[k0 + 1];
    const float* r0 = sW + (PR_WIN + p0) * PPROW + aoff;
    const float* s0 = sW + (PR_WSH + p0) * PPROW + aoff;
#pragma unroll
    for (int mt = 0; mt < 4; ++mt) {
      v2f a = *(const v2f*)(r0 + mt * 32);
      accH[mt] = __builtin_amdgcn_wmma_f32_16x16x4_f32(
          false, a, false, bf, (short)0, accH[mt], false, false);
      v2f s = *(const v2f*)(s0 + mt * 32);
      accS[mt] = __builtin_amdgcn_wmma_f32_16x16x4_f32(
          false, s, false, bf, (short)0, accS[mt], false, false);
    }
  }
  // ---- GEMM1b: K in [64,192) from sobel-grad channels -------------------
#pragma unroll 4
  for (int kb = FCH; kb < C3F; kb += 4) {
    const int k0 = kb + 2 * half;
    const int p0 = (kb >> 1) + half;
    v2f bf;
    bf.x = (pGrad[(size_t)(kb - FCH) * HW]      - sMean[k0])     * sIstd[k0];
    bf.y = (pGrad[(size_t)(kb - FCH) * HW + HW] - sMean[k0 + 1]) * sIstd[k0 + 1];
    const float* r0 = sW + (PR_WIN + p0) * PPROW + aoff;
    const float* s0 = sW + (PR_WSH + p0) * PPROW + aoff;
#pragma unroll
    for (int mt = 0; mt < 4; ++mt) {
      v2f a = *(const v2f*)(r0 + mt * 32);
      accH[mt] = __builtin_amdgcn_wmma_f32_16x16x4_f32(
          false, a, false, bf, (short)0, accH[mt], false, false);
      v2f s = *(const v2f*)(s0 + mt * 32);
      accS[mt] = __builtin_amdgcn_wmma_f32_16x16x4_f32(
          false, s, false, bf, (short)0, accS[mt], false, false);
    }
  }

  // ---- bias + lrelu, stash h1 into wave-private LDS ---------------------
  const float* bIn  = sW + BI_IN;
  const float* bMid = sW + BI_MID;
  const float* bOut = sW + BI_OUT;
  const float* bSh  = sW + BI_SH;
  float* myH = sH + wave * (FCH * 16);
#pragma unroll
  for (int mt = 0; mt < 4; ++mt)
#pragma unroll
    for (int r = 0; r < 8; ++r) {
      int m = mt * 16 + r + 8 * half;
      myH[m * 16 + ll] = lrelu(accH[mt][r] + bIn[m]);
    }

  // ---- GEMM2: h2 = lrelu(W_mid * h1 + b_mid)  (K = 64) ------------------
  v8f accM[4] = {};
#pragma unroll 4
  for (int kb = 0; kb < FCH; kb += 4) {
    const int k0 = kb + 2 * half;
    const int p0 = (kb >> 1) + half;
    v2f bf;
    bf.x = myH[k0 * 16 + ll];
    bf.y = myH[(k0 + 1) * 16 + ll];
    const float* r0 = sW + (PR_WMID + p0) * PPROW + aoff;
#pragma unroll
    for (int mt = 0; mt < 4; ++mt) {
      v2f a = *(const v2f*)(r0 + mt * 32);
      accM[mt] = __builtin_amdgcn_wmma_f32_16x16x4_f32(
          false, a, false, bf, (short)0, accM[mt], false, false);
    }
  }
#pragma unroll
  for (int mt = 0; mt < 4; ++mt)
#pragma unroll
    for (int r = 0; r < 8; ++r) {
      int m = mt * 16 + r + 8 * half;
      myH[m * 16 + ll] = lrelu(accM[mt][r] + bMid[m]);
    }

  // ---- GEMM3: o = W_out * h2 + b_out  (K = 64) --------------------------
  v8f accO[4] = {};
#pragma unroll 4
  for (int kb = 0; kb < FCH; kb += 4) {
    const int k0 = kb + 2 * half;
    const int p0 = (kb >> 1) + half;
    v2f bf;
    bf.x = myH[k0 * 16 + ll];
    bf.y = myH[(k0 + 1) * 16 + ll];
    const float* r0 = sW + (PR_WOUT + p0) * PPROW + aoff;
#pragma unroll
    for (int mt = 0; mt < 4; ++mt) {
      v2f a = *(const v2f*)(r0 + mt * 32);
      accO[mt] = __builtin_amdgcn_wmma_f32_16x16x4_f32(
          false, a, false, bf, (short)0, accO[mt], false, false);
    }
  }

  // ---- state update: out += lf * (o + b_out + sc + b_sh) ----------------
  float lf = lf_ptr[0];
  lf = fminf(fmaxf(lf, 0.001f), 1000.f);
#pragma unroll
  for (int mt = 0; mt < 4; ++mt)
#pragma unroll
    for (int r = 0; r < 8; ++r) {
      int m = mt * 16 + r + 8 * half;
      float o = accO[mt][r] + bOut[m];
      float s = accS[mt][r] + bSh[m];
      size_t idx = identBase + (size_t)m * HW + pix;
      state_out[idx] = state_in[idx] + lf * (o + s);
    }
}

// ---------------------------------------------------------------------------
// Kernel 3: generic 3x3 direct conv (Cin=64), LDS row tiles, channel chunks.
// flags: bit0 = leaky-relu, bit1 = clamp[-1,1].  resid!=null => out=resid+conv
// ---------------------------------------------------------------------------
__global__ __launch_bounds__(256) void conv3x3_kernel(
    const float* __restrict__ in,     // [B][64][128][128]
    const float* __restrict__ w,      // [Cout][64][3][3]
    const float* __restrict__ bias,   // [Cout]
    const float* __restrict__ resid,  // [B][Cout][128][128] or null
    float* __restrict__ out,          // [B][Cout][128][128]
    int Cout, int flags)
{
  const int row = blockIdx.x;
  const int b   = blockIdx.y;
  const int tid = threadIdx.x;
  __shared__ float tile[16 * 3 * 130];   // 16 in-channels, 3 rows, 130 cols
  float acc[32];
#pragma unroll
  for (int i = 0; i < 32; ++i) acc[i] = 0.f;
  const int nOut = Cout * HH;

  for (int cc = 0; cc < FCH; cc += 16) {
    __syncthreads();
    for (int i = tid; i < 16 * 3 * 130; i += 256) {
      int c = i / (3 * 130);
      int rem = i % (3 * 130);
      int rr = rem / 130, col = rem % 130;
      int gr = row + rr - 1, gc = col - 1;
      float v = 0.f;
      if (gr >= 0 && gr < HH && gc >= 0 && gc < HH)
        v = in[(((size_t)b * FCH + cc + c) * HH + gr) * HH + gc];
      tile[i] = v;
    }
    __syncthreads();
    int slot = 0;
    for (int t = tid; t < nOut; t += 256, ++slot) {
      int o = t >> 7, px = t & 127;
      float a = acc[slot];
      const float* wp = w + ((size_t)o * FCH + cc) * 9;
      for (int c = 0; c < 16; ++c) {
        const float* wc = wp + c * 9;
        const float* tc = tile + c * (3 * 130);
#pragma unroll
        for (int ky = 0; ky < 3; ++ky)
#pragma unroll
          for (int kx = 0; kx < 3; ++kx)
            a += tc[ky * 130 + px + kx] * wc[ky * 3 + kx];
      }
      acc[slot] = a;
    }
  }

  int slot = 0;
  for (int t = tid; t < nOut; t += 256, ++slot) {
    int o = t >> 7, px = t & 127;
    float a = acc[slot] + bias[o];
    if (flags & 1) a = lrelu(a);
    size_t idx = (((size_t)b * Cout + o) * HH + row) * HH + px;
    if (resid) a += resid[idx];
    if (flags & 2) a = fminf(fmaxf(a, -1.f), 1.f);
    out[idx] = a;
  }
}

// ---------------------------------------------------------------------------
extern "C" void kernel_launch(void* const* d_in, const int* in_sizes, int n_in,
                              void* d_out, int out_size, void* d_ws, size_t ws_size,
                              hipStream_t stream) {
  const float* lat     = (const float*)d_in[0];
  const float* ca_init = (const float*)d_in[1];
  const float* leak    = (const float*)d_in[2];
  const float* hyper_w = (const float*)d_in[3];
  const float* hyper_b = (const float*)d_in[4];
  const float* res_w1  = (const float*)d_in[5];
  const float* res_b1  = (const float*)d_in[6];
  const float* res_w2  = (const float*)d_in[7];
  const float* res_b2  = (const float*)d_in[8];
  const float* img_w   = (const float*)d_in[9];
  const float* img_b   = (const float*)d_in[10];
  float* out = (float*)d_out;

  char* ws = (char*)d_ws;
  size_t off = 0;
  auto alloc = [&](size_t bytes) -> void* {
    void* p = ws + off;
    off += (bytes + 255) & ~(size_t)255;
    return p;
  };
  const size_t stateBytes = (size_t)NB * FCH * HW * sizeof(float);   // 16 MB
  float* state0  = (float*)alloc(stateBytes);
  float* state1  = (float*)alloc(stateBytes);
  float* grad    = (float*)alloc((size_t)NB * 2 * FCH * HW * sizeof(float));
  float* meanArr = (float*)alloc((size_t)NB * C3F * sizeof(float));
  float* istdArr = (float*)alloc((size_t)NB * C3F * sizeof(float));
  float* wt      = (float*)alloc((size_t)NB * HYP * sizeof(float));
  float* tmpH    = (float*)alloc(stateBytes);
  float* out2    = (float*)alloc(stateBytes);

  hipMemcpyAsync(state0, ca_init, stateBytes, hipMemcpyDeviceToDevice, stream);
  hyper_kernel<<<(HYP + 255) / 256, 256, 0, stream>>>(lat, hyper_w, hyper_b, wt);

  const size_t dynLds = (size_t)L_TOTAL * sizeof(float);   // 199,424 bytes
  float* s_in = state0;
  float* s_out = state1;
  for (int it = 0; it < 16; ++it) {
    sobel_stats_kernel<<<NB * FCH, 256, 0, stream>>>(s_in, grad, meanArr, istdArr);
    dyna_step_kernel<<<dim3(HH, NB), 256, dynLds, stream>>>(s_in, grad, meanArr, istdArr,
                                                            wt, leak, s_out);
    float* t = s_in; s_in = s_out; s_out = t;
  }

  conv3x3_kernel<<<dim3(HH, NB), 256, 0, stream>>>(s_in, res_w1, res_b1, nullptr, tmpH, FCH, 1);
  conv3x3_kernel<<<dim3(HH, NB), 256, 0, stream>>>(tmpH, res_w2, res_b2, s_in, out2, FCH, 0);
  conv3x3_kernel<<<dim3(HH, NB), 256, 0, stream>>>(out2, img_w, img_b, nullptr, out, 3, 2);
}